// BWENet_9740985827669
// MI455X (gfx1250) — compile-verified
//
#include <hip/hip_runtime.h>
#include <math.h>

typedef float v2f __attribute__((ext_vector_type(2)));
typedef float v8f __attribute__((ext_vector_type(8)));

#define BATCH 8
#define CD    128
#define FDIM  84
#define TT    400      // feature frames
#define NF    800      // cf frames after 2x transposed conv
#define N32   128000   // samples at 32k rate (per batch)
#define N48   192000   // samples at 48k rate (per batch)
#define KTAP  15

#define CDIV(a,b) (((a)+(b)-1)/(b))

// _FRAC rows 0, 8, 4 (scaled by 1/32768 at use site)
static __device__ __constant__ float FRAC_A[8] = {189.f,-600.f,617.f,30567.f,2996.f,-1375.f,425.f,-46.f};
static __device__ __constant__ float FRAC_B[8] = {-103.f,896.f,-3487.f,11950.f,26341.f,-3350.f,529.f,-4.f};
static __device__ __constant__ float FRAC_C[8] = {-48.f,758.f,-3956.f,23973.f,15143.f,-3957.f,967.f,-107.f};

// ---------------------------------------------------------------------------
// Build the two reversed 128-tap HQ2X impulse responses (module constants in
// the reference, computed here on-device in double precision, single thread),
// and zero the GEMM zero-row scratch (predicate-free N-edge handling).
// ---------------------------------------------------------------------------
__global__ void init_const_k(float* we, float* wo, float* zrow) {
  if (threadIdx.x != 0 || blockIdx.x != 0) return;
  const double ce[3] = {1746.0/65536.0, 14986.0/65536.0, (39083.0-65536.0)/65536.0};
  const double co[3] = {6854.0/65536.0, 25769.0/65536.0, (55542.0-65536.0)/65536.0};
  for (int tab = 0; tab < 2; ++tab) {
    const double* c = tab ? co : ce;
    float* w = tab ? wo : we;
    double s0 = 0.0, s1 = 0.0, s2 = 0.0, xin = 1.0;
    for (int i = 0; i < 128; ++i) {
      double Y = xin - s0, X = Y * c[0]; double t1 = s0 + X; s0 = xin + X;
      Y = t1 - s1; X = Y * c[1];         double t2 = s1 + X; s1 = t1 + X;
      Y = t2 - s2; X = Y * (1.0 + c[2]); double t3 = s2 + X; s2 = t2 + X;
      w[127 - i] = (float)t3;            // store reversed (matches y[::-1])
      xin = 0.0;
    }
  }
  for (int i = 0; i < CD; ++i) zrow[i] = 0.f;
}

// ---------------------------------------------------------------------------
// hq_2x_up: y[b, 2t+ph] = sum_j w_ph[j] * x[b, t+j-127]
// ---------------------------------------------------------------------------
__global__ void hq2x_k(const float* __restrict__ x, const float* __restrict__ we,
                       const float* __restrict__ wo, float* __restrict__ y) {
  int idx = blockIdx.x * blockDim.x + threadIdx.x;
  if (idx >= BATCH * N32) return;
  int n = idx % N32, b = idx / N32;
  int t = n >> 1;
  const float* w  = (n & 1) ? wo : we;
  const float* xb = x + (size_t)b * (N32 / 2);
  float acc = 0.f;
  int j0 = 127 - t; if (j0 < 0) j0 = 0;
  for (int j = j0; j < 128; ++j) acc += w[j] * xb[t + j - 127];
  y[idx] = acc;
}

// ---------------------------------------------------------------------------
// feature_net convs (thread per output, causal)
// ---------------------------------------------------------------------------
__global__ void conv1_k(const float* __restrict__ feat, const float* __restrict__ w,
                        const float* __restrict__ bias, float* __restrict__ out) {
  int idx = blockIdx.x * blockDim.x + threadIdx.x;     // (b*TT+t)*CD + co
  if (idx >= BATCH * TT * CD) return;
  int co = idx % CD, t = (idx / CD) % TT, b = idx / (CD * TT);
  float acc = bias[co];
  for (int r = 0; r < 3; ++r) {
    int tt = t - 2 + r; if (tt < 0) continue;
    const float* fr = feat + ((size_t)b * TT + tt) * FDIM;
    const float* wr = w + (size_t)co * FDIM * 3 + r;
    for (int ci = 0; ci < FDIM; ++ci) acc += fr[ci] * wr[ci * 3];
  }
  out[idx] = tanhf(acc);
}

__global__ void conv2_k(const float* __restrict__ in, const float* __restrict__ w,
                        const float* __restrict__ bias, float* __restrict__ out) {
  int idx = blockIdx.x * blockDim.x + threadIdx.x;
  if (idx >= BATCH * TT * CD) return;
  int co = idx % CD, t = (idx / CD) % TT, b = idx / (CD * TT);
  float acc = bias[co];
  for (int r = 0; r < 3; ++r) {
    int tt = t - 2 + r; if (tt < 0) continue;
    const float* fr = in + ((size_t)b * TT + tt) * CD;
    const float* wr = w + (size_t)co * CD * 3 + r;
    for (int ci = 0; ci < CD; ++ci) acc += fr[ci] * wr[ci * 3];
  }
  out[idx] = tanhf(acc);
}

// transposed conv (k=2, stride 2): xs[b, 2t+r, o] = tanh(b[o] + sum_i c2[b,t,i]*W[i,o,r])
__global__ void tconv_k(const float* __restrict__ in, const float* __restrict__ w,
                        const float* __restrict__ bias, float* __restrict__ out) {
  int idx = blockIdx.x * blockDim.x + threadIdx.x;     // (b*NF+s)*CD + o
  if (idx >= BATCH * NF * CD) return;
  int o = idx % CD, s = (idx / CD) % NF, b = idx / (CD * NF);
  int t = s >> 1, r = s & 1;
  float acc = bias[o];
  const float* fr = in + ((size_t)b * TT + t) * CD;
  for (int i = 0; i < CD; ++i) acc += fr[i] * w[((size_t)i * CD + o) * 2 + r];
  out[idx] = tanhf(acc);
}

// ---------------------------------------------------------------------------
// WMMA f32 TN GEMM: C[M,N] = A[M,K] * W[N,K]^T + bias[N]
// One wave -> one 16x16 tile, K consumed 4 at a time via V_WMMA_F32_16X16X4_F32.
// A frag (16x4): lane = (k/2)*16 + m, vgpr = k&1  ->  lane loads A[m, k0+2*kh .. +1]
// B frag (4x16): lane = (k/2)*16 + n, vgpr = k&1  ->  lane loads W[n, k0+2*kh .. +1]
// C/D (16x16): vgpr v: lanes0-15 -> M=v, lanes16-31 -> M=v+8; N = lane&15
// Out-of-range N rows read a zeroed scratch row instead of branching, so the
// inner loop is predicate-free (EXEC all-1s, as WMMA requires) and contains
// just 2x global_load_b64 + v_wmma per K-step.
// ---------------------------------------------------------------------------
__global__ void gemm_tn_wmma(const float* __restrict__ A, const float* __restrict__ W,
                             const float* __restrict__ bias, const float* __restrict__ zrow,
                             float* __restrict__ C, int M, int N, int K, int ldc) {
  int wave = threadIdx.x >> 5;
  int lane = threadIdx.x & 31;
  int ntiles = (N + 15) >> 4;
  int mtiles = M >> 4;                       // M is always a multiple of 16 here
  int tile = blockIdx.x * (blockDim.x >> 5) + wave;
  if (tile >= mtiles * ntiles) return;
  int tm = tile / ntiles, tn = tile - tm * ntiles;
  int mBase = tm << 4, nBase = tn << 4;
  int half = lane & 15, kh = lane >> 4;

  const float* arow = A + (size_t)(mBase + half) * K + 2 * kh;
  int nrow = nBase + half;
  bool wok = nrow < N;
  const float* wrow = (wok ? (W + (size_t)nrow * K) : zrow) + 2 * kh;

  v8f acc = {0.f, 0.f, 0.f, 0.f, 0.f, 0.f, 0.f, 0.f};
#pragma unroll 4
  for (int k0 = 0; k0 < K; k0 += 4) {
    v2f a  = *(const v2f*)(arow + k0);
    v2f bv = *(const v2f*)(wrow + k0);
    acc = __builtin_amdgcn_wmma_f32_16x16x4_f32(false, a, false, bv,
                                                (short)0, acc, false, false);
  }
  float bb = wok ? bias[nrow] : 0.f;
#pragma unroll
  for (int v = 0; v < 8; ++v) {
    int m = mBase + v + kh * 8;
    if (nrow < ldc) C[(size_t)m * ldc + nrow] = acc[v] + bb;
  }
}

// ---------------------------------------------------------------------------
// GRU scan: one block per batch element, 384 threads (12 waves), h in LDS.
// gi precomputed (WMMA GEMM). gh = h @ Whh^T + bhh each step.
// ---------------------------------------------------------------------------
__global__ void gru_scan_k(const float* __restrict__ gi, const float* __restrict__ whh,
                           const float* __restrict__ bhh, float* __restrict__ cf) {
  int b = blockIdx.x, tid = threadIdx.x;
  __shared__ float h[CD];
  __shared__ float gh[3 * CD];
  if (tid < CD) h[tid] = 0.f;
  float bh = bhh[tid];
  const float* wr = whh + (size_t)tid * CD;
  __syncthreads();
  for (int s = 0; s < NF; ++s) {
    float acc = bh;
    for (int k = 0; k < CD; ++k) acc += h[k] * wr[k];
    gh[tid] = acc;
    __syncthreads();
    if (tid < CD) {
      const float* gir = gi + ((size_t)b * NF + s) * (3 * CD);
      float r = 1.f / (1.f + expf(-(gir[tid]          + gh[tid])));
      float z = 1.f / (1.f + expf(-(gir[CD + tid]     + gh[CD + tid])));
      float n = tanhf(gir[2 * CD + tid] + r * gh[2 * CD + tid]);
      float hn = (1.f - z) * n + z * h[tid];
      h[tid] = hn;
      cf[((size_t)b * NF + s) * CD + tid] = hn;
    }
    __syncthreads();
  }
}

// ---------------------------------------------------------------------------
// Adaptive-conv kernel normalization + gain.
// kraw: [B*NF][32] (padded GEMM out), kfin: [B*NF][oc][ick]
// g = exp(12*LOGDB * tanh(cf . gw[o] + gb[o]))   (gb_const = 0)
// ---------------------------------------------------------------------------
__global__ void norm_gain_k(const float* __restrict__ kraw, const float* __restrict__ cf,
                            const float* __restrict__ gw, const float* __restrict__ gb,
                            float* __restrict__ kfin, int oc, int ick) {
  int idx = blockIdx.x * blockDim.x + threadIdx.x;
  if (idx >= BATCH * NF * oc) return;
  int o = idx % oc, m = idx / oc;
  const float* kr = kraw + (size_t)m * 32 + o * ick;
  float ss = 0.f;
  for (int i = 0; i < ick; ++i) { float v = kr[i]; ss += v * v; }
  float inv = 1.f / (1e-6f + sqrtf(ss));
  const float* cfr = cf + (size_t)m * CD;
  const float* gwr = gw + (size_t)o * CD;
  float d = gb[o];
  for (int i = 0; i < CD; ++i) d += cfr[i] * gwr[i];
  float g = expf(1.3815510557964275f * tanhf(d));   // 12 dB * LOGDB
  float sc = inv * g;
  float* ko = kfin + ((size_t)m * oc + o) * ick;
  for (int i = 0; i < ick; ++i) ko[i] = kr[i] * sc;
}

// ---------------------------------------------------------------------------
// Adaptive FIR: Y[b,f,o,l] = sum_{c,j} K[b,f,o,c,j] * x[b,c, f*FS + j + l - (k-1)]
// ---------------------------------------------------------------------------
__global__ void afir_k(const float* __restrict__ x, const float* __restrict__ kfin,
                       float* __restrict__ Y, int ic, int oc, int FS, int OV, int N) {
  int L = FS + OV;
  int idx = blockIdx.x * blockDim.x + threadIdx.x;
  int total = BATCH * NF * oc * L;
  if (idx >= total) return;
  int l = idx % L;
  int o = (idx / L) % oc;
  int f = (idx / (L * oc)) % NF;
  int b = idx / (L * oc * NF);
  const float* kk = kfin + (((size_t)b * NF + f) * oc + o) * ic * KTAP;
  int base = f * FS + l - (KTAP - 1);
  float acc = 0.f;
  for (int c = 0; c < ic; ++c) {
    const float* xc = x + ((size_t)b * ic + c) * N;
    const float* kc = kk + c * KTAP;
#pragma unroll
    for (int j = 0; j < KTAP; ++j) {
      int xi = base + j;
      float xv = (xi >= 0 && xi < N) ? xc[xi] : 0.f;
      acc += kc[j] * xv;
    }
  }
  Y[idx] = acc;
}

// overlap-add with half-sine crossfade
__global__ void oadd_k(const float* __restrict__ Y, float* __restrict__ out,
                       int oc, int FS, int OV, int N) {
  int L = FS + OV;
  int idx = blockIdx.x * blockDim.x + threadIdx.x;
  int total = BATCH * oc * N;
  if (idx >= total) return;
  int n = idx % N, o = (idx / N) % oc, b = idx / (N * oc);
  int f = n / FS, s = n - f * FS;
  const float* Yf = Y + (((size_t)b * NF + f) * oc + o) * L;
  float v;
  if (s < OV) {
    const float PI = 3.14159265358979323846f;
    float wu = sinf(0.5f * PI * (s + 0.5f) / OV);
    float wd = sinf(0.5f * PI * ((OV - 1 - s) + 0.5f) / OV);
    float prev = (f > 0) ? Y[(((size_t)b * NF + (f - 1)) * oc + o) * L + FS + s] : 0.f;
    v = Yf[s] * wu + prev * wd;
  } else {
    v = Yf[s];
  }
  out[idx] = v;
}

// ---------------------------------------------------------------------------
// 3/2 interpolation: y48[b,3t+ph] from x32 (stride-2 polyphase, left pad 8)
// ---------------------------------------------------------------------------
__global__ void interp_k(const float* __restrict__ x32, float* __restrict__ y48) {
  int idx = blockIdx.x * blockDim.x + threadIdx.x;
  if (idx >= BATCH * N48) return;
  int m = idx % N48, b = idx / N48;
  int t = m / 3, ph = m - 3 * t;
  const float* w = (ph == 0) ? FRAC_A : (ph == 1) ? FRAC_B : FRAC_C;
  int base = 2 * t + (ph == 2 ? 1 : 0) - 8;
  const float* xb = x32 + (size_t)b * N32;
  float acc = 0.f;
#pragma unroll
  for (int j = 0; j < 8; ++j) {
    int xi = base + j;
    if (xi >= 0) acc += w[j] * xb[xi];
  }
  y48[idx] = acc * (1.f / 32768.f);
}

// ---------------------------------------------------------------------------
// td_shaper part 1: per-frame log envelope, tenv[b,f,0:20]=e-avg, tenv[b,f,20]=avg
// ---------------------------------------------------------------------------
__global__ void env_k(const float* __restrict__ xch, float* __restrict__ tenv,
                      int FS, int apk, int bstride) {
  int bf = blockIdx.x;                        // b*NF + f
  int f = bf % NF, b = bf / NF;
  int m1 = FS / apk;                          // 20
  int tid = threadIdx.x;
  __shared__ float e[32];
  __shared__ float avg_s;
  float ev = 0.f;
  if (tid < m1) {
    const float* xb = xch + (size_t)b * bstride + (size_t)f * FS + tid * apk;
    float s = 0.f;
    for (int i = 0; i < apk; ++i) s += fabsf(xb[i]);
    ev = logf(s / apk + 1.52587890625e-5f);   // + 2^-16
    e[tid] = ev;
  }
  __syncthreads();
  if (tid == 0) {
    float a = 0.f;
    for (int i = 0; i < m1; ++i) a += e[i];
    avg_s = a / m1;
  }
  __syncthreads();
  float* tv = tenv + ((size_t)b * NF + f) * 21;
  if (tid < m1) tv[tid] = ev - avg_s;
  if (tid == m1) tv[m1] = avg_s;
}

// td_shaper alpha stage 1: leaky(conv_k2(cf) + conv_k2(tenv))  -> a1[b,f,s]
__global__ void alpha1_k(const float* __restrict__ cf, const float* __restrict__ tenv,
                         const float* __restrict__ wf, const float* __restrict__ bf_,
                         const float* __restrict__ wt, const float* __restrict__ bt_,
                         float* __restrict__ a1, int fs) {
  int idx = blockIdx.x * blockDim.x + threadIdx.x;
  if (idx >= BATCH * NF * fs) return;
  int s = idx % fs, f = (idx / fs) % NF, b = idx / (fs * NF);
  float acc = bf_[s] + bt_[s];
  for (int r = 0; r < 2; ++r) {
    int ff = f - 1 + r; if (ff < 0) continue;
    const float* cfr = cf + ((size_t)b * NF + ff) * CD;
    const float* wr  = wf + (size_t)s * CD * 2 + r;
    for (int c = 0; c < CD; ++c) acc += cfr[c] * wr[c * 2];
    const float* tr  = tenv + ((size_t)b * NF + ff) * 21;
    const float* wtr = wt + (size_t)s * 21 * 2 + r;
    for (int e = 0; e < 21; ++e) acc += tr[e] * wtr[e * 2];
  }
  a1[idx] = (acc >= 0.f) ? acc : 0.2f * acc;
}

// td_shaper alpha stage 2 + apply (in place on channel-1 samples; safe: the
// channel buffer is fully regenerated by oadd_k earlier in the same call)
__global__ void alpha2_k(const float* __restrict__ a1, const float* __restrict__ w2,
                         const float* __restrict__ b2, float* __restrict__ xch,
                         int fs, int bstride) {
  int idx = blockIdx.x * blockDim.x + threadIdx.x;
  if (idx >= BATCH * NF * fs) return;
  int s = idx % fs, f = (idx / fs) % NF, b = idx / (fs * NF);
  float acc = b2[s];
  for (int r = 0; r < 2; ++r) {
    int ff = f - 1 + r; if (ff < 0) continue;
    const float* ar = a1 + ((size_t)b * NF + ff) * fs;
    const float* wr = w2 + (size_t)s * fs * 2 + r;
    for (int s2 = 0; s2 < fs; ++s2) acc += ar[s2] * wr[s2 * 2];
  }
  size_t xi = (size_t)b * bstride + (size_t)f * fs + s;   // fs == frame_size
  xch[xi] = expf(acc) * xch[xi];
}

// ---------------------------------------------------------------------------
extern "C" void kernel_launch(void* const* d_in, const int* in_sizes, int n_in,
                              void* d_out, int out_size, void* d_ws, size_t ws_size,
                              hipStream_t stream) {
  (void)in_sizes; (void)n_in; (void)out_size; (void)ws_size;
  const float* x        = (const float*)d_in[0];
  const float* features = (const float*)d_in[1];
  const float* c1_w = (const float*)d_in[2];  const float* c1_b = (const float*)d_in[3];
  const float* c2_w = (const float*)d_in[4];  const float* c2_b = (const float*)d_in[5];
  const float* tc_w = (const float*)d_in[6];  const float* tc_b = (const float*)d_in[7];
  const float* wih  = (const float*)d_in[8];  const float* whh  = (const float*)d_in[9];
  const float* bih  = (const float*)d_in[10]; const float* bhh  = (const float*)d_in[11];
  const float* af_kw[4] = {(const float*)d_in[12], (const float*)d_in[16],
                           (const float*)d_in[20], (const float*)d_in[24]};
  const float* af_kb[4] = {(const float*)d_in[13], (const float*)d_in[17],
                           (const float*)d_in[21], (const float*)d_in[25]};
  const float* af_gw[4] = {(const float*)d_in[14], (const float*)d_in[18],
                           (const float*)d_in[22], (const float*)d_in[26]};
  const float* af_gb[4] = {(const float*)d_in[15], (const float*)d_in[19],
                           (const float*)d_in[23], (const float*)d_in[27]};
  const float* sh_a1fw[2] = {(const float*)d_in[28], (const float*)d_in[34]};
  const float* sh_a1fb[2] = {(const float*)d_in[29], (const float*)d_in[35]};
  const float* sh_a1tw[2] = {(const float*)d_in[30], (const float*)d_in[36]};
  const float* sh_a1tb[2] = {(const float*)d_in[31], (const float*)d_in[37]};
  const float* sh_a2w[2]  = {(const float*)d_in[32], (const float*)d_in[38]};
  const float* sh_a2b[2]  = {(const float*)d_in[33], (const float*)d_in[39]};

  float* ws = (float*)d_ws;
  size_t off = 0;
  float* w_even = ws + off; off += 128;
  float* w_odd  = ws + off; off += 128;
  float* zrow   = ws + off; off += 128;
  float* c1buf  = ws + off; off += (size_t)BATCH * TT * CD;       // 409600
  float* c2buf  = ws + off; off += (size_t)BATCH * TT * CD;
  float* xsbuf  = ws + off; off += (size_t)BATCH * NF * CD;       // 819200
  float* gibuf  = ws + off; off += (size_t)BATCH * NF * 3 * CD;   // 2457600
  float* cfbuf  = ws + off; off += (size_t)BATCH * NF * CD;
  float* y32up  = ws + off; off += (size_t)BATCH * N32;           // 1024000
  float* kraw   = ws + off; off += (size_t)BATCH * NF * 32;       // 204800
  float* kfin   = ws + off; off += (size_t)BATCH * NF * 32;
  float* buf232 = ws + off; off += (size_t)BATCH * 2 * N32;       // 2048000
  float* Ybuf   = ws + off; off += (size_t)BATCH * NF * 2 * (240 + 120); // 4608000 (max stage)
  float* y32f   = ws + off; off += (size_t)BATCH * N32;
  float* y48    = ws + off; off += (size_t)BATCH * N48;
  float* buf248 = ws + off; off += (size_t)BATCH * 2 * N48;       // 3072000
  float* tenv   = ws + off; off += (size_t)BATCH * NF * 21;
  float* a1buf  = ws + off; off += (size_t)BATCH * NF * 240;      // max fs

  const int TB = 256;
  const int M  = BATCH * NF;            // 6400 GEMM rows

  // --- constants + base 2x upsample (independent of feature chain) ---
  init_const_k<<<1, 1, 0, stream>>>(w_even, w_odd, zrow);
  hq2x_k<<<CDIV(BATCH * N32, TB), TB, 0, stream>>>(x, w_even, w_odd, y32up);

  // --- feature_net ---
  conv1_k<<<CDIV(BATCH * TT * CD, TB), TB, 0, stream>>>(features, c1_w, c1_b, c1buf);
  conv2_k<<<CDIV(BATCH * TT * CD, TB), TB, 0, stream>>>(c1buf, c2_w, c2_b, c2buf);
  tconv_k<<<CDIV(BATCH * NF * CD, TB), TB, 0, stream>>>(c2buf, tc_w, tc_b, xsbuf);
  {
    int tiles = (M / 16) * CDIV(3 * CD, 16);                     // 400*24
    gemm_tn_wmma<<<CDIV(tiles, 4), 128, 0, stream>>>(xsbuf, wih, bih, zrow, gibuf,
                                                     M, 3 * CD, CD, 3 * CD);
  }
  gru_scan_k<<<BATCH, 3 * CD, 0, stream>>>(gibuf, whh, bhh, cfbuf);

  // --- adaptive-stage kernel generator: WMMA GEMM + normalize/gain ---
  auto kgen = [&](int st, int oc, int ick) {
    int tiles = (M / 16) * CDIV(30, 16);                          // 400*2
    gemm_tn_wmma<<<CDIV(tiles, 4), 128, 0, stream>>>(cfbuf, af_kw[st], af_kb[st], zrow,
                                                     kraw, M, 30, CD, 32);
    norm_gain_k<<<CDIV(M * oc, TB), TB, 0, stream>>>(kraw, cfbuf, af_gw[st], af_gb[st],
                                                     kfin, oc, ick);
  };

  // --- 32k chain: af1 (1->2), shaper1 on ch1, af2 (2->1) ---
  kgen(0, 2, 15);
  afir_k<<<CDIV(BATCH * NF * 2 * 240, TB), TB, 0, stream>>>(y32up, kfin, Ybuf, 1, 2, 160, 80, N32);
  oadd_k<<<CDIV(BATCH * 2 * N32, TB), TB, 0, stream>>>(Ybuf, buf232, 2, 160, 80, N32);

  env_k<<<BATCH * NF, 32, 0, stream>>>(buf232 + N32, tenv, 160, 8, 2 * N32);
  alpha1_k<<<CDIV(BATCH * NF * 160, TB), TB, 0, stream>>>(cfbuf, tenv, sh_a1fw[0], sh_a1fb[0],
                                                          sh_a1tw[0], sh_a1tb[0], a1buf, 160);
  alpha2_k<<<CDIV(BATCH * NF * 160, TB), TB, 0, stream>>>(a1buf, sh_a2w[0], sh_a2b[0],
                                                          buf232 + N32, 160, 2 * N32);

  kgen(1, 1, 30);
  afir_k<<<CDIV(BATCH * NF * 1 * 240, TB), TB, 0, stream>>>(buf232, kfin, Ybuf, 2, 1, 160, 80, N32);
  oadd_k<<<CDIV(BATCH * 1 * N32, TB), TB, 0, stream>>>(Ybuf, y32f, 1, 160, 80, N32);

  // --- 3/2 resample to 48k ---
  interp_k<<<CDIV(BATCH * N48, TB), TB, 0, stream>>>(y32f, y48);

  // --- 48k chain: af3 (1->2), shaper2 on ch1, af4 (2->1) -> d_out ---
  kgen(2, 2, 15);
  afir_k<<<CDIV(BATCH * NF * 2 * 360, TB), TB, 0, stream>>>(y48, kfin, Ybuf, 1, 2, 240, 120, N48);
  oadd_k<<<CDIV(BATCH * 2 * N48, TB), TB, 0, stream>>>(Ybuf, buf248, 2, 240, 120, N48);

  env_k<<<BATCH * NF, 32, 0, stream>>>(buf248 + N48, tenv, 240, 12, 2 * N48);
  alpha1_k<<<CDIV(BATCH * NF * 240, TB), TB, 0, stream>>>(cfbuf, tenv, sh_a1fw[1], sh_a1fb[1],
                                                          sh_a1tw[1], sh_a1tb[1], a1buf, 240);
  alpha2_k<<<CDIV(BATCH * NF * 240, TB), TB, 0, stream>>>(a1buf, sh_a2w[1], sh_a2b[1],
                                                          buf248 + N48, 240, 2 * N48);

  kgen(3, 1, 30);
  afir_k<<<CDIV(BATCH * NF * 1 * 360, TB), TB, 0, stream>>>(buf248, kfin, Ybuf, 2, 1, 240, 120, N48);
  oadd_k<<<CDIV(BATCH * 1 * N48, TB), TB, 0, stream>>>(Ybuf, (float*)d_out, 1, 240, 120, N48);
}